// CausalGraphLayer_70360154243501
// MI455X (gfx1250) — compile-verified
//
#include <hip/hip_runtime.h>
#include <math.h>

typedef __attribute__((ext_vector_type(2))) float v2f;
typedef __attribute__((ext_vector_type(8))) float v8f;

// ---------------------------------------------------------------------------
// Kernel 1: S = rowsoftmax(C) ; M = W @ S ; bS = b @ S   (single block, tiny)
// ---------------------------------------------------------------------------
__global__ __launch_bounds__(256)
void prep_kernel(const float* __restrict__ W, const float* __restrict__ b,
                 const float* __restrict__ C, float* __restrict__ Mo,
                 float* __restrict__ bS)
{
    __shared__ __align__(16) float S[64 * 64];
    __shared__ __align__(16) float sW[64 * 64];
    const int tid = threadIdx.x;

    for (int i = tid; i < 4096; i += 256) sW[i] = W[i];

    if (tid < 64) {
        float m = -INFINITY;
        for (int j = 0; j < 64; ++j) m = fmaxf(m, C[tid * 64 + j]);
        float sum = 0.0f;
        for (int j = 0; j < 64; ++j) {
            float ev = expf(C[tid * 64 + j] - m);
            S[tid * 64 + j] = ev;
            sum += ev;
        }
        float inv = 1.0f / sum;
        for (int j = 0; j < 64; ++j) S[tid * 64 + j] *= inv;
    }
    __syncthreads();

    // M = W @ S  (each thread: 16 outputs of 64-MAC dot products)
    for (int idx = tid; idx < 4096; idx += 256) {
        int i = idx >> 6, j = idx & 63;
        float acc = 0.0f;
        for (int k = 0; k < 64; ++k) acc = fmaf(sW[i * 64 + k], S[k * 64 + j], acc);
        Mo[idx] = acc;
    }
    if (tid < 64) {
        float acc = 0.0f;
        for (int k = 0; k < 64; ++k) acc = fmaf(b[k], S[k * 64 + tid], acc);
        bS[tid] = acc;
    }
}

// ---------------------------------------------------------------------------
// Degree pipeline: deg=1 (self loop) ; +1 per in-edge ; dinv = rsqrt(deg)
// ---------------------------------------------------------------------------
__global__ void deg_init(float* deg, int n)
{
    int i = blockIdx.x * blockDim.x + threadIdx.x;
    if (i < n) deg[i] = 1.0f;
}

__global__ void deg_count(const int* __restrict__ ei, float* deg, int E)
{
    int e = blockIdx.x * blockDim.x + threadIdx.x;
    if (e < E) atomicAdd(&deg[ei[E + e]], 1.0f);  // dst = edge_index[1][e]
}

__global__ void dinv_kernel(float* d, int n)
{
    int i = blockIdx.x * blockDim.x + threadIdx.x;
    if (i < n) d[i] = rsqrtf(d[i]);  // deg >= 1 always (self loop)
}

// ---------------------------------------------------------------------------
// Kernel: y = x @ M  via V_WMMA_F32_16X16X4_F32 (wave32, one 16x16 tile/wave)
// Block: 256 thr = 8 waves -> 32 rows x 64 cols per block. K = 64 -> 16 WMMAs.
// ---------------------------------------------------------------------------
__global__ __launch_bounds__(256)
void gemm_wmma_f32(const float* __restrict__ x, const float* __restrict__ M,
                   float* __restrict__ y, int nrows)
{
    __shared__ __align__(16) float sA[32 * 64];  // x tile, 8 KB
    __shared__ __align__(16) float sB[64 * 64];  // M,      16 KB

    const int tid = threadIdx.x;
    const int rowbase = blockIdx.x * 32;

    // Stage M (coalesced float4)
    for (int i = tid * 4; i < 4096; i += 256 * 4)
        *(float4*)&sB[i] = *(const float4*)&M[i];

    // Stage 32 rows of x (coalesced float4), zero-fill past nrows
    for (int i = tid * 4; i < 2048; i += 256 * 4) {
        int row = rowbase + (i >> 6);
        float4 v = make_float4(0.f, 0.f, 0.f, 0.f);
        if (row < nrows) v = *(const float4*)&x[(size_t)row * 64 + (i & 63)];
        *(float4*)&sA[i] = v;
    }
    __syncthreads();

    const int wave = tid >> 5;       // 0..7
    const int lane = tid & 31;
    const int rblk = wave >> 2;      // 0..1  (row block of 16)
    const int cblk = wave & 3;       // 0..3  (col block of 16)
    const int r    = lane & 15;      // row (A) / col (B,D) index within tile
    const int hi   = lane >> 4;      // lane half: selects K sub-pairs

    const float* Arow = &sA[(rblk * 16 + r) * 64];
    const int    bcol = cblk * 16 + r;

    v8f acc = {};  // C starts at zero, chained through K
#pragma unroll
    for (int k0 = 0; k0 < 64; k0 += 4) {
        // A 16x4 f32: VGPR0 = K {0 | 2}, VGPR1 = K {1 | 3} per lane half
        v2f a, bfr;
        a.x = Arow[k0 + 2 * hi + 0];
        a.y = Arow[k0 + 2 * hi + 1];
        // B 4x16 f32: row striped across lanes; same K-pair split per half
        bfr.x = sB[(k0 + 2 * hi + 0) * 64 + bcol];
        bfr.y = sB[(k0 + 2 * hi + 1) * 64 + bcol];
        acc = __builtin_amdgcn_wmma_f32_16x16x4_f32(
            /*neg_a=*/false, a, /*neg_b=*/false, bfr,
            /*c_mod=*/(short)0, acc, /*reuse_a=*/false, /*reuse_b=*/false);
    }

    // D layout: VGPR v -> M = v + 8*hi, N = r
    const int outrow = rowbase + rblk * 16 + hi * 8;
    const int outcol = cblk * 16 + r;
#pragma unroll
    for (int v = 0; v < 8; ++v) {
        int rr = outrow + v;
        if (rr < nrows) y[(size_t)rr * 64 + outcol] = acc[v];
    }
}

// ---------------------------------------------------------------------------
// out[i,:] = dinv[i]^2 * y[i,:] + bS   (initializes poisoned d_out)
// ---------------------------------------------------------------------------
__global__ __launch_bounds__(256)
void self_init(const float* __restrict__ y, const float* __restrict__ dinv,
               const float* __restrict__ bS, float* __restrict__ out, int n)
{
    int t = blockIdx.x * blockDim.x + threadIdx.x;  // 16 threads per node
    int i = t >> 4;
    int c = (t & 15) * 4;
    if (i >= n) return;
    float w = dinv[i] * dinv[i];
    float4 v  = *(const float4*)&y[(size_t)i * 64 + c];
    float4 bb = *(const float4*)&bS[c];
    float4 o  = make_float4(w * v.x + bb.x, w * v.y + bb.y,
                            w * v.z + bb.z, w * v.w + bb.w);
    *(float4*)&out[(size_t)i * 64 + c] = o;
}

// ---------------------------------------------------------------------------
// Edge scatter: out[dst,:] += dinv[src]*dinv[dst] * y[src,:]
// 16 lanes/edge, float4 gather + 4 atomic f32 adds per lane
// ---------------------------------------------------------------------------
__global__ __launch_bounds__(256)
void edge_scatter(const int* __restrict__ ei, const float* __restrict__ dinv,
                  const float* __restrict__ y, float* __restrict__ out, int E)
{
    int t = blockIdx.x * blockDim.x + threadIdx.x;
    int e = t >> 4;
    int c = (t & 15) * 4;
    if (e >= E) return;
    int s = ei[e];        // src = edge_index[0][e]
    int d = ei[E + e];    // dst = edge_index[1][e]
    float w = dinv[s] * dinv[d];
    float4 v = *(const float4*)&y[(size_t)s * 64 + c];
    float* o = &out[(size_t)d * 64 + c];
    atomicAdd(o + 0, w * v.x);
    atomicAdd(o + 1, w * v.y);
    atomicAdd(o + 2, w * v.z);
    atomicAdd(o + 3, w * v.w);
}

// ---------------------------------------------------------------------------
extern "C" void kernel_launch(void* const* d_in, const int* in_sizes, int n_in,
                              void* d_out, int out_size, void* d_ws, size_t ws_size,
                              hipStream_t stream)
{
    const float* x  = (const float*)d_in[0];
    const int*   ei = (const int*)d_in[1];
    const float* W  = (const float*)d_in[2];
    const float* b  = (const float*)d_in[3];
    const float* Ca = (const float*)d_in[4];
    // d_in[5] = L (==1), handled implicitly (single layer)

    const int N = in_sizes[0] / 64;
    const int E = in_sizes[1] / 2;

    float* ws   = (float*)d_ws;
    float* M    = ws;              // 4096 floats
    float* bS   = ws + 4096;       // 64
    float* dinv = ws + 4160;       // N   (deg -> dinv in place)
    float* y    = ws + 4160 + N;   // N*64 (16B-aligned: (4160+N)*4 % 16 == 0)
    float* out  = (float*)d_out;

    prep_kernel <<<1, 256, 0, stream>>>(W, b, Ca, M, bS);
    deg_init    <<<(N + 255) / 256, 256, 0, stream>>>(dinv, N);
    deg_count   <<<(E + 255) / 256, 256, 0, stream>>>(ei, dinv, E);
    dinv_kernel <<<(N + 255) / 256, 256, 0, stream>>>(dinv, N);

    gemm_wmma_f32<<<(N + 31) / 32, 256, 0, stream>>>(x, M, y, N);

    self_init   <<<(N * 16 + 255) / 256, 256, 0, stream>>>(y, dinv, bS, out, N);

    long long escatter_threads = (long long)E * 16;
    edge_scatter<<<(int)((escatter_threads + 255) / 256), 256, 0, stream>>>(
        ei, dinv, y, out, E);
}